// kuramoto_angular_closed_loop_19636590477633
// MI455X (gfx1250) — compile-verified
//
#include <hip/hip_runtime.h>

typedef __attribute__((ext_vector_type(2))) float v2f;
typedef __attribute__((ext_vector_type(8))) float v8f;

// D = A(16x4) x B(4x16) + C, full f32, one wave32.
static __device__ __forceinline__ v8f wmma_f32_16x16x4(v2f a, v2f b, v8f c) {
  return __builtin_amdgcn_wmma_f32_16x16x4_f32(
      /*neg_a=*/false, a, /*neg_b=*/false, b,
      /*c_mod=*/(short)0, c, /*reuse_a=*/false, /*reuse_b=*/false);
}

static __device__ __forceinline__ float fast_tanh(float x) {
#if __has_builtin(__builtin_amdgcn_tanhf)
  return __builtin_amdgcn_tanhf(x);   // CDNA5 V_TANH (TRANS op)
#else
  return tanhf(x);
#endif
}

__global__ __launch_bounds__(32)
void kuramoto_angular_closed_loop_kernel(const float* __restrict__ X,    // x, 12
                                         const float* __restrict__ Kp,   // K, 16
                                         const float* __restrict__ bp,   // b, 4
                                         const float* __restrict__ Gp,   // G, 16
                                         const float* __restrict__ Ap,   // adj, 16
                                         float* __restrict__ out)        // 12
{
  const int lane = threadIdx.x;        // 0..31 (one wave32)
  const int m    = lane & 15;          // A row / B-D column index
  const int hi   = lane >> 4;          // which K-half this lane holds
  const int k0   = 2 * hi;
  const int k1   = 2 * hi + 1;
  const int rb   = m & 3;              // row within a 4-row block

  // ---------- per-lane distributed copies: lane l holds flat element l&15 ----
  const float kv  = Kp[m];             // K  (row-major, 16)
  const float gv  = Gp[m];             // G
  const float agv = Ap[m] * gv;        // adj*G

  // ---------- uniform scalar copies (redundant in every lane) ----------
  float a[4], v[4], xiv[4], b[4], adjv[16];
#pragma unroll
  for (int i = 0; i < 4; ++i) { a[i] = X[i]; v[i] = X[4 + i]; xiv[i] = X[8 + i]; b[i] = bp[i]; }
#pragma unroll
  for (int i = 0; i < 16; ++i) adjv[i] = Ap[i];

  // ---------- branchless WMMA operand gathers via cross-lane shuffles ----------
  const int idx0 = rb * 4 + k0;        // A block-row element (row rb, col k0/k1)
  const int idx1 = rb * 4 + k1;
  const float K0  = __shfl(kv,  idx0, 32), K1  = __shfl(kv,  idx1, 32);
  const float AG0 = __shfl(agv, idx0, 32), AG1 = __shfl(agv, idx1, 32);
  const float G0  = __shfl(gv,  idx0, 32), G1  = __shfl(gv,  idx1, 32);

  // ---------- WMMA #1 : A1 = [K ; adj*G ; G ; 0], B1 cols {0:xi, 1:x2, 8..11:G^T}
  v2f a1;
  a1.x = (m < 4) ? K0 : (m < 8) ? AG0 : (m < 12) ? G0 : 0.0f;
  a1.y = (m < 4) ? K1 : (m < 8) ? AG1 : (m < 12) ? G1 : 0.0f;
  v2f b1;
  b1.x = (m == 0) ? (hi ? xiv[2] : xiv[0])
       : (m == 1) ? (hi ? v[2]   : v[0])
       : (m >= 8 && m < 12) ? G0 : 0.0f;   // B1[k][8+j] = G[j][k] -> flat j*4+k = idx0
  b1.y = (m == 0) ? (hi ? xiv[3] : xiv[1])
       : (m == 1) ? (hi ? v[3]   : v[1])
       : (m >= 8 && m < 12) ? G1 : 0.0f;
  v8f zc = {};
  v8f d1 = wmma_f32_16x16x4(a1, b1, zc);

  // Extract: Kxi[i]=D1[i][0] (VGPR i, lane 0); AGx2[i]=D1[4+i][1] (VGPR 4+i, lane 1);
  //          GG[i][j]=D1[8+i][8+j] (VGPR i, lane 24+j).
  float Kxi[4], AGx2[4], GG[4][4];
#pragma unroll
  for (int i = 0; i < 4; ++i) {
    Kxi[i]  = __shfl(d1[i],     0, 32);
    AGx2[i] = __shfl(d1[4 + i], 1, 32);
#pragma unroll
    for (int j = 0; j < 4; ++j) GG[i][j] = __shfl(d1[i], 24 + j, 32);
  }

  float t2[4];
#pragma unroll
  for (int i = 0; i < 4; ++i) t2[i] = fast_tanh(Kxi[i] + b[i]);

  // ---------- WMMA #2 : dH = K^T @ t2 ----------
  const float KT0 = __shfl(kv, k0 * 4 + rb, 32);   // K^T[m][k] = K[k][m]
  const float KT1 = __shfl(kv, k1 * 4 + rb, 32);
  v2f a2;
  a2.x = (m < 4) ? KT0 : 0.0f;
  a2.y = (m < 4) ? KT1 : 0.0f;
  v2f b2;
  b2.x = (m == 0) ? (hi ? t2[2] : t2[0]) : 0.0f;
  b2.y = (m == 0) ? (hi ? t2[3] : t2[1]) : 0.0f;
  v8f d2 = wmma_f32_16x16x4(a2, b2, zc);

  float dH[4];
#pragma unroll
  for (int i = 0; i < 4; ++i) dH[i] = __shfl(d2[i], 0, 32);

  // ---------- WMMA #3 : u = -(adj*G)^T @ dH ----------
  const float AGT0 = __shfl(agv, k0 * 4 + rb, 32); // (AG)^T[m][k] = AG[k][m]
  const float AGT1 = __shfl(agv, k1 * 4 + rb, 32);
  v2f a3;
  a3.x = (m < 4) ? AGT0 : 0.0f;
  a3.y = (m < 4) ? AGT1 : 0.0f;
  v2f b3;
  b3.x = (m == 0) ? (hi ? dH[2] : dH[0]) : 0.0f;
  b3.y = (m == 0) ? (hi ? dH[3] : dH[1]) : 0.0f;
  v8f d3 = wmma_f32_16x16x4(a3, b3, zc);

  float u[4];
#pragma unroll
  for (int i = 0; i < 4; ++i) u[i] = -__shfl(d3[i], 0, 32);

  // ---------- interactions[j] = sum_i adj[i][j] * cos(a_j - a_i) * (v_j - v_i)
  float inter[4];
#pragma unroll
  for (int j = 0; j < 4; ++j) {
    float s = 0.0f;
#pragma unroll
    for (int i = 0; i < 4; ++i)
      s += adjv[i * 4 + j] * __cosf(a[j] - a[i]) * (v[j] - v[i]);
    inter[j] = s;
  }

  float dxdt[4];
#pragma unroll
  for (int i = 0; i < 4; ++i) dxdt[i] = 1.25f * u[i] * inter[i];

  // ---------- gamma = 0.1 * lambda_max(G G^T) via power iteration (SPD) ----------
  float y0 = 0.5f, y1 = 0.5f, y2 = 0.5f, y3 = 0.5f;  // ||y|| = 1
  float lam = 0.0f;
  for (int it = 0; it < 50; ++it) {
    float z0 = GG[0][0]*y0 + GG[0][1]*y1 + GG[0][2]*y2 + GG[0][3]*y3;
    float z1 = GG[1][0]*y0 + GG[1][1]*y1 + GG[1][2]*y2 + GG[1][3]*y3;
    float z2 = GG[2][0]*y0 + GG[2][1]*y1 + GG[2][2]*y2 + GG[2][3]*y3;
    float z3 = GG[3][0]*y0 + GG[3][1]*y1 + GG[3][2]*y2 + GG[3][3]*y3;
    lam = y0*z0 + y1*z1 + y2*z2 + y3*z3;            // Rayleigh quotient (y normalized)
    float inv = 1.0f / sqrtf(z0*z0 + z1*z1 + z2*z2 + z3*z3);
    y0 = z0 * inv; y1 = z1 * inv; y2 = z2 * inv; y3 = z3 * inv;
  }
  const float gamma = 0.1f * lam;

  // ---------- dxi_dt = (J - gamma I) @ dH + (adj*G) @ x2 ----------
  float JdH[4] = { -dH[1], dH[0], -dH[3], dH[2] };
  float dxi[4];
#pragma unroll
  for (int i = 0; i < 4; ++i) dxi[i] = JdH[i] - gamma * dH[i] + AGx2[i];

  // ---------- write 12 outputs from lane 0 ----------
  if (lane == 0) {
#pragma unroll
    for (int i = 0; i < 4; ++i) out[i] = v[i];
#pragma unroll
    for (int i = 0; i < 4; ++i) out[4 + i] = dxdt[i];
#pragma unroll
    for (int i = 0; i < 4; ++i) out[8 + i] = dxi[i];
  }
}

extern "C" void kernel_launch(void* const* d_in, const int* in_sizes, int n_in,
                              void* d_out, int out_size, void* d_ws, size_t ws_size,
                              hipStream_t stream) {
  // setup_inputs order: t(1), x(12), K(16), b(4), G(16), adj(16); t unused.
  const float* x   = (const float*)d_in[1];
  const float* K   = (const float*)d_in[2];
  const float* b   = (const float*)d_in[3];
  const float* G   = (const float*)d_in[4];
  const float* adj = (const float*)d_in[5];
  float* out = (float*)d_out;
  kuramoto_angular_closed_loop_kernel<<<dim3(1), dim3(32), 0, stream>>>(x, K, b, G, adj, out);
}